// ChargeEquilibrationBlock_13872744366326
// MI455X (gfx1250) — compile-verified
//
#include <hip/hip_runtime.h>
#include <hip/hip_bf16.h>
#include <math.h>

typedef float v2f __attribute__((ext_vector_type(2)));
typedef float v8f __attribute__((ext_vector_type(8)));

#define NATOM    128
#define NB       16           // LU panel width
#define NPANEL   8            // 128 / 16
#define LDA      133          // 133 mod 64 = 5, gcd(5,64)=1 -> conflict-free row stride
#define NTHREADS 128          // 4 waves (wave32)

// One workgroup per molecule. A (129x130 incl. rhs col 129) lives entirely in
// LDS (CDNA5: 320KB/WGP -> 4 resident ~70KB workgroups; matrix never touches
// HBM). Build: pairwise Gram matrix via V_WMMA_F32_16X16X4_F32. Solve: blocked
// LU (NB=16); the O(n^3) Schur update A22 -= L21*U12 runs on WMMA with K
// chunked 4 at a time (multipliers stored negated so D = (-L)*U + C).
__global__ __launch_bounds__(NTHREADS, 1)
void qeq_fused_kernel(const float* __restrict__ eneg,
                      const float* __restrict__ positions,
                      const float* __restrict__ node_attrs,
                      const float* __restrict__ hardness,
                      const float* __restrict__ total_charge,
                      const int*   __restrict__ atomic_numbers,
                      float* __restrict__ out)
{
    __shared__ float sA[129 * LDA];                      // 68628 B
    __shared__ float spx[NATOM], spy[NATOM], spz[NATOM];
    __shared__ float snorm[NATOM], ssigma[NATOM], sdiag[NATOM];

    const int t    = threadIdx.x;
    const int mol  = blockIdx.x;
    const int gi   = mol * NATOM + t;
    const int wave = t >> 5;
    const int lane = t & 31;
    const int lhi  = lane >> 4;      // 0: K={0,1} half, 1: K={2,3} half
    const int llo  = lane & 15;

    // ---- prologue: per-atom scalars + border/rhs ----
    {
        const float px = positions[gi * 3 + 0];
        const float py = positions[gi * 3 + 1];
        const float pz = positions[gi * 3 + 2];
        const int   Z  = atomic_numbers[gi];
        const float r  = 0.3f + 0.02f * (float)Z;        // covalent radii table

        float best = node_attrs[gi * 10];                // first-argmax, 10 channels
        int   bidx = 0;
        #pragma unroll
        for (int e = 1; e < 10; ++e) {
            const float v = node_attrs[gi * 10 + e];
            if (v > best) { best = v; bidx = e; }
        }

        spx[t] = px; spy[t] = py; spz[t] = pz;
        snorm[t]  = px * px + py * py + pz * pz;
        ssigma[t] = r * r;
        sdiag[t]  = hardness[bidx] + 1.0f / (1.7724538509055159f * r);

        sA[t * LDA + 128]  = 1.0f;                       // ones column
        sA[t * LDA + 129]  = -eneg[gi];                  // rhs column
        sA[128 * LDA + t]  = 1.0f;                       // ones row
        if (t == 0) {
            sA[128 * LDA + 128] = 1.0f;
            sA[128 * LDA + 129] = total_charge[mol];
        }
    }
    __syncthreads();

    // ---- build A[0:128,0:128]: Gram matrix via WMMA, then erf kernel ----
    for (int tt = wave; tt < 64; tt += 4) {              // uniform per wave: EXEC all-1s
        const int i0 = (tt >> 3) * 16;
        const int j0 = (tt & 7) * 16;

        v2f a, b;                                        // f32 16x4 operand layout
        if (lhi == 0) {
            a.x = spx[i0 + llo]; a.y = spy[i0 + llo];
            b.x = spx[j0 + llo]; b.y = spy[j0 + llo];
        } else {
            a.x = spz[i0 + llo]; a.y = 0.0f;
            b.x = spz[j0 + llo]; b.y = 0.0f;
        }

        v8f acc = {};
        acc = __builtin_amdgcn_wmma_f32_16x16x4_f32(false, a, false, b,
                                                    (short)0, acc, false, false);

        const int j = j0 + llo;
        #pragma unroll
        for (int v = 0; v < 8; ++v) {
            const int i = i0 + v + 8 * lhi;
            float val;
            if (i == j) {
                val = sdiag[i];
            } else {
                const float sq = snorm[i] + snorm[j] - 2.0f * acc[v];
                const float d  = sqrtf(fmaxf(sq, 1e-12f));
                const float g  = sqrtf(ssigma[i] + ssigma[j]);
                val = erff(d / (1.4142135623730951f * g)) / d;
            }
            sA[i * LDA + j] = val;
        }
    }
    __syncthreads();

    // ---- blocked LU, no pivoting (diag = hardness + 1/(sqrt(pi) r) dominates) ----
    for (int kb = 0; kb < NPANEL; ++kb) {
        const int kb0   = kb * NB;
        const int ncols = 130 - (kb0 + NB);              // trailing cols incl 128,129

        // Phase 1: panel factorization (cols kb0..kb0+15, rows below pivot).
        // Multipliers stored NEGATED for the WMMA update.
        for (int k = 0; k < NB; ++k) {
            const int K = kb0 + k;
            const float invp = 1.0f / sA[K * LDA + K];   // LDS broadcast
            const int i = K + 1 + t;                     // one row per thread
            if (i <= 128) {
                const float f = sA[i * LDA + K] * invp;
                sA[i * LDA + K] = -f;
                for (int j = K + 1; j <= kb0 + NB - 1; ++j)
                    sA[i * LDA + j] -= f * sA[K * LDA + j];
            }
            __syncthreads();
        }

        // Phase 2: U12 = L11^{-1} * A12, one trailing column per thread.
        if (t < ncols) {
            const int j = kb0 + NB + t;
            float col[NB];
            #pragma unroll
            for (int k = 0; k < NB; ++k) col[k] = sA[(kb0 + k) * LDA + j];
            #pragma unroll
            for (int k = 1; k < NB; ++k) {
                float acc = col[k];
                #pragma unroll
                for (int m = 0; m < k; ++m)              // L11 stored as -mult
                    acc += sA[(kb0 + k) * LDA + (kb0 + m)] * col[m];
                col[k] = acc;
            }
            #pragma unroll
            for (int k = 0; k < NB; ++k) sA[(kb0 + k) * LDA + j] = col[k];
        }
        __syncthreads();

        // Phase 3: A22 += (-L21)*U12 on WMMA, full 16x16 tiles (rows/cols <=127).
        const int nt16   = NPANEL - 1 - kb;
        const int ntiles = nt16 * nt16;
        for (int tt = wave; tt < ntiles; tt += 4) {
            const int i0 = kb0 + NB + (tt / nt16) * NB;
            const int j0 = kb0 + NB + (tt % nt16) * NB;

            v8f acc;                                     // C tile
            #pragma unroll
            for (int v = 0; v < 8; ++v)
                acc[v] = sA[(i0 + v + 8 * lhi) * LDA + j0 + llo];

            #pragma unroll
            for (int c = 0; c < 4; ++c) {                // K = 16 in chunks of 4
                const int ka = kb0 + 4 * c + 2 * lhi;
                v2f a, b;
                a.x = sA[(i0 + llo) * LDA + ka];
                a.y = sA[(i0 + llo) * LDA + ka + 1];
                b.x = sA[ka       * LDA + j0 + llo];
                b.y = sA[(ka + 1) * LDA + j0 + llo];
                acc = __builtin_amdgcn_wmma_f32_16x16x4_f32(false, a, false, b,
                                                            (short)0, acc, false, false);
            }

            #pragma unroll
            for (int v = 0; v < 8; ++v)
                sA[(i0 + v + 8 * lhi) * LDA + j0 + llo] = acc[v];
        }

        // Ragged leftovers (write-disjoint from the tiles -> same barrier region):
        // row 128 across all trailing columns
        if (t < ncols) {
            const int j = kb0 + NB + t;
            float acc = sA[128 * LDA + j];
            #pragma unroll
            for (int k = 0; k < NB; ++k)
                acc += sA[128 * LDA + kb0 + k] * sA[(kb0 + k) * LDA + j];
            sA[128 * LDA + j] = acc;
        }
        // columns 128/129 for the full-tile rows (<=127)
        const int nbelow = 128 - (kb0 + NB);
        for (int p = t; p < nbelow * 2; p += NTHREADS) {
            const int i = kb0 + NB + (p >> 1);
            const int c = 128 + (p & 1);
            float acc = sA[i * LDA + c];
            #pragma unroll
            for (int k = 0; k < NB; ++k)
                acc += sA[i * LDA + kb0 + k] * sA[(kb0 + k) * LDA + c];
            sA[i * LDA + c] = acc;
        }
        __syncthreads();
    }

    // ---- back substitution (upper triangle + rhs col only) ----
    float myx = 0.0f;
    for (int k = 128; k >= 0; --k) {
        const float xk = sA[k * LDA + 129] / sA[k * LDA + k];
        if (t == k) myx = xk;                            // k==128 multiplier not kept
        if (t < k)  sA[t * LDA + 129] -= sA[t * LDA + k] * xk;
        __syncthreads();
    }

    out[gi] = myx;
}

extern "C" void kernel_launch(void* const* d_in, const int* in_sizes, int n_in,
                              void* d_out, int out_size, void* d_ws, size_t ws_size,
                              hipStream_t stream) {
    const float* eneg           = (const float*)d_in[0];
    const float* positions      = (const float*)d_in[1];
    const float* node_attrs     = (const float*)d_in[2];
    const float* hardness       = (const float*)d_in[3];
    const float* total_charge   = (const float*)d_in[4];
    // d_in[5] = batch (implied by layout: molecule b owns atoms [b*128, b*128+128))
    const int*   atomic_numbers = (const int*)d_in[6];
    float* out = (float*)d_out;

    const int B = in_sizes[4];   // number of molecules (1024)
    qeq_fused_kernel<<<B, NTHREADS, 0, stream>>>(eneg, positions, node_attrs,
                                                 hardness, total_charge,
                                                 atomic_numbers, out);
    (void)d_ws; (void)ws_size; (void)out_size; (void)n_in;
}